// GAT_58299886075957
// MI455X (gfx1250) — compile-verified
//
#include <hip/hip_runtime.h>
#include <math.h>

typedef __attribute__((ext_vector_type(2))) float v2f;
typedef __attribute__((ext_vector_type(8))) float v8f;

#define HDIM 128

// ---------------------------------------------------------------------------
// monotone float -> u32 key so atomicMax(u32) implements float max
__device__ __forceinline__ unsigned fkey(float f) {
    unsigned u = __float_as_uint(f);
    return (u & 0x80000000u) ? ~u : (u | 0x80000000u);
}
__device__ __forceinline__ float unfkey(unsigned k) {
    return __uint_as_float((k & 0x80000000u) ? (k ^ 0x80000000u) : ~k);
}
__device__ __forceinline__ float leaky(float x) { return x > 0.f ? x : 0.2f * x; }

// ---------------------------------------------------------------------------
// Big fp32 WMMA GEMM, Ncols fixed to 128 at compile time.
// C[M,128] = op_pre(A[M,KK]) * B[KK,128], epilogue op_post.
// Block = 4 waves; each wave computes 32 rows (2 M-tiles) x 128 cols so every
// B fragment read from LDS feeds two V_WMMA_F32_16X16X4_F32.
// B is staged through LDS in 64-row K-chunks with the CDNA5 async
// global->LDS engine (GLOBAL_LOAD_ASYNC_TO_LDS_B128 / ASYNCcnt).
// PRE:  0 = none, 1 = A + preb[k], 2 = relu(A + preb[k])
// POST: 0 = none, 1 = relu(x + postb[n])
template <int KK, int PRE, int POST>
__global__ void __launch_bounds__(128)
gemm_wmma_lds(const float* __restrict__ A, const float* __restrict__ Bm,
              const float* __restrict__ preb, const float* __restrict__ postb,
              float* __restrict__ C, int M) {
    constexpr int NC   = 128;      // output columns (compile time)
    constexpr int KCH  = 64;       // K rows staged per LDS chunk
    constexpr int LDB  = NC + 4;   // padded LDS row stride (floats)
    __shared__ float bs[KCH * LDB];

    const int tid  = threadIdx.x;
    const int lane = tid & 31;
    const int wave = tid >> 5;
    const int l16  = lane & 15;
    const int hi   = lane >> 4;                    // half-wave -> K pair (ISA 7.12.2)
    const int m0   = (blockIdx.x * 4 + wave) * 32; // 2 tiles of 16 rows

    int rowA[2];
#pragma unroll
    for (int mi = 0; mi < 2; ++mi) {
        int r = m0 + mi * 16 + l16;
        rowA[mi] = r < M ? r : (M - 1);            // clamp (no divergent loads)
    }

    v8f acc[2][8];
#pragma unroll
    for (int mi = 0; mi < 2; ++mi)
#pragma unroll
        for (int t = 0; t < 8; ++t)
#pragma unroll
            for (int r = 0; r < 8; ++r) acc[mi][t][r] = 0.0f;

    const unsigned lds_base = (unsigned)(size_t)(void*)&bs[0];

    for (int kc = 0; kc < KK; kc += KCH) {
        __syncthreads();   // previous chunk fully consumed before overwrite
        // ---- async global -> LDS stage of B rows [kc, kc+KCH) -------------
#pragma unroll
        for (int it = 0; it < (KCH * (NC / 4)) / 128; ++it) {
            int idx = tid + it * 128;
            int row = idx >> 5;                    // 32 float4 per row
            int c4  = idx & 31;
            unsigned       loff  = lds_base + (unsigned)(row * LDB + c4 * 4) * 4u;
            unsigned long long g = (unsigned long long)(Bm + (size_t)(kc + row) * NC + c4 * 4);
            asm volatile("global_load_async_to_lds_b128 %0, %1, off"
                         :: "v"(loff), "v"(g) : "memory");
        }
        asm volatile("s_wait_asynccnt 0x0" ::: "memory");
        __syncthreads();

        // ---- compute on the staged chunk ----------------------------------
        for (int k0 = 0; k0 < KCH; k0 += 4) {
            const int kg = kc + k0 + 2 * hi;       // global K for A / preb
            const int kl = k0 + 2 * hi;            // local K inside LDS chunk
            v2f a[2];
#pragma unroll
            for (int mi = 0; mi < 2; ++mi) {
                v2f av = *(const v2f*)(A + (size_t)rowA[mi] * KK + kg);
                float ax = av[0], ay = av[1];
                if (PRE >= 1) { ax += preb[kg]; ay += preb[kg + 1]; }
                if (PRE == 2) { ax = ax > 0.f ? ax : 0.f; ay = ay > 0.f ? ay : 0.f; }
                a[mi][0] = ax; a[mi][1] = ay;
            }
#pragma unroll
            for (int t = 0; t < 8; ++t) {
                const int col = t * 16 + l16;
                v2f b;
                b[0] = bs[kl * LDB + col];
                b[1] = bs[(kl + 1) * LDB + col];
                acc[0][t] = __builtin_amdgcn_wmma_f32_16x16x4_f32(
                    false, a[0], false, b, (short)0, acc[0][t], false, false);
                acc[1][t] = __builtin_amdgcn_wmma_f32_16x16x4_f32(
                    false, a[1], false, b, (short)0, acc[1][t], false, false);
            }
        }
    }

    // ---- epilogue + store --------------------------------------------------
#pragma unroll
    for (int mi = 0; mi < 2; ++mi)
#pragma unroll
        for (int t = 0; t < 8; ++t) {
            const int col = t * 16 + l16;
#pragma unroll
            for (int r = 0; r < 8; ++r) {
                const int orow = m0 + mi * 16 + r + 8 * hi; // VGPR r -> M=r / r+8
                float v = acc[mi][t][r];
                if (POST >= 1) v += postb[col];
                if (POST == 1) v = v > 0.f ? v : 0.f;
                if (orow < M) C[(size_t)orow * NC + col] = v;
            }
        }
}

// ---------------------------------------------------------------------------
// Final narrow GEMM: C[M,Ncols] = A[M,K] * B[K,Ncols], Ncols <= 32 (here 20),
// sigmoid epilogue. Masked columns use clamped addresses + v_cndmask selects,
// never EXEC divergence, so the wave stays converged into the WMMAs.
__global__ void __launch_bounds__(128)
gemm_wmma_small(const float* __restrict__ A, const float* __restrict__ Bm,
                const float* __restrict__ postb, float* __restrict__ C,
                int M, int K, int Ncols) {
    const int lane = threadIdx.x & 31;
    const int wave = threadIdx.x >> 5;
    const int mt   = blockIdx.x * 4 + wave;
    if (mt * 16 >= M) return;                      // wave-uniform exit
    const int l16 = lane & 15;
    const int hi  = lane >> 4;
    const int row = mt * 16 + l16;

    v8f acc[2];
#pragma unroll
    for (int t = 0; t < 2; ++t)
#pragma unroll
        for (int r = 0; r < 8; ++r) acc[t][r] = 0.0f;

    for (int k0 = 0; k0 < K; k0 += 4) {
        const int ka = k0 + 2 * hi;
        v2f a = *(const v2f*)(A + (size_t)row * K + ka);
#pragma unroll
        for (int t = 0; t < 2; ++t) {
            const int col  = t * 16 + l16;
            const int colc = col < Ncols ? col : (Ncols - 1);
            float bx = Bm[(size_t)ka * Ncols + colc];
            float by = Bm[(size_t)(ka + 1) * Ncols + colc];
            v2f b;
            b[0] = col < Ncols ? bx : 0.f;         // v_cndmask, no EXEC change
            b[1] = col < Ncols ? by : 0.f;
            acc[t] = __builtin_amdgcn_wmma_f32_16x16x4_f32(
                false, a, false, b, (short)0, acc[t], false, false);
        }
    }
#pragma unroll
    for (int t = 0; t < 2; ++t) {
        const int col = t * 16 + l16;
#pragma unroll
        for (int r = 0; r < 8; ++r) {
            const int orow = mt * 16 + r + 8 * hi;
            float v = acc[t][r] + postb[col < Ncols ? col : 0];
            v = 1.0f / (1.0f + __expf(-v));
            if (col < Ncols && orow < M) C[(size_t)orow * Ncols + col] = v;
        }
    }
}

// ---------------------------------------------------------------------------
// per-node attention dots: as[i] = h[i].a_src ; ad[i] = h[i].a_dst  (wave/node)
__global__ void node_alphas(const float* __restrict__ h,
                            const float* __restrict__ avs,
                            const float* __restrict__ avd,
                            float* __restrict__ outs, float* __restrict__ outd,
                            int n) {
    const int wid  = (int)((blockIdx.x * blockDim.x + threadIdx.x) >> 5);
    const int lane = threadIdx.x & 31;
    if (wid >= n) return;
    float4 hv = ((const float4*)(h + (size_t)wid * HDIM))[lane];
    float4 s4 = ((const float4*)avs)[lane];
    float4 d4 = ((const float4*)avd)[lane];
    float ps = hv.x * s4.x + hv.y * s4.y + hv.z * s4.z + hv.w * s4.w;
    float pd = hv.x * d4.x + hv.y * d4.y + hv.z * d4.z + hv.w * d4.w;
#pragma unroll
    for (int off = 16; off > 0; off >>= 1) {
        ps += __shfl_xor(ps, off, 32);
        pd += __shfl_xor(pd, off, 32);
    }
    if (lane == 0) { outs[wid] = ps; outd[wid] = pd; }
}

// ---------------------------------------------------------------------------
__global__ void gat_init(float* __restrict__ agg, unsigned* __restrict__ maxkey,
                         float* __restrict__ denom, int n_nodes) {
    const int i = blockIdx.x * blockDim.x + threadIdx.x;
    if (i < n_nodes * HDIM) agg[i] = 0.0f;
    if (i < n_nodes) { maxkey[i] = 0u; denom[i] = 0.0f; }
}

__global__ void edge_logit_max(const int* __restrict__ src, const int* __restrict__ dst,
                               const float* __restrict__ as, const float* __restrict__ ad,
                               unsigned* __restrict__ maxkey, int Eb, int n) {
    const int e = blockIdx.x * blockDim.x + threadIdx.x;
    if (e >= Eb + n) return;
    int s, d;
    if (e < Eb) { s = src[e]; d = dst[e]; } else { s = d = e - Eb; }
    atomicMax(&maxkey[d], fkey(leaky(as[s] + ad[d])));
}

__global__ void edge_expsum(const int* __restrict__ src, const int* __restrict__ dst,
                            const float* __restrict__ as, const float* __restrict__ ad,
                            const unsigned* __restrict__ maxkey,
                            float* __restrict__ wexp, float* __restrict__ denom,
                            int Eb, int n) {
    const int e = blockIdx.x * blockDim.x + threadIdx.x;
    if (e >= Eb + n) return;
    int s, d;
    if (e < Eb) { s = src[e]; d = dst[e]; } else { s = d = e - Eb; }
    float w = __expf(leaky(as[s] + ad[d]) - unfkey(maxkey[d]));
    wexp[e] = w;
    atomicAdd(&denom[d], w);
}

// wave per edge: agg[dst] += (wexp/denom[dst]) * h[src]   (float4 per lane)
__global__ void edge_aggregate(const int* __restrict__ src, const int* __restrict__ dst,
                               const float* __restrict__ h,
                               const float* __restrict__ wexp,
                               const float* __restrict__ denom,
                               float* __restrict__ agg, int Eb, int n) {
    const int wid  = (int)((blockIdx.x * blockDim.x + threadIdx.x) >> 5);
    const int lane = threadIdx.x & 31;
    if (wid >= Eb + n) return;
    int s, d;
    if (wid < Eb) { s = src[wid]; d = dst[wid]; } else { s = d = wid - Eb; }
    const float coef = wexp[wid] / denom[d];
    float4 hv = ((const float4*)(h + (size_t)s * HDIM))[lane];
    float* ap = agg + (size_t)d * HDIM + lane * 4;
    atomicAdd(ap + 0, coef * hv.x);
    atomicAdd(ap + 1, coef * hv.y);
    atomicAdd(ap + 2, coef * hv.z);
    atomicAdd(ap + 3, coef * hv.w);
}

// ---------------------------------------------------------------------------
extern "C" void kernel_launch(void* const* d_in, const int* in_sizes, int n_in,
                              void* d_out, int out_size, void* d_ws, size_t ws_size,
                              hipStream_t stream) {
    const float* x   = (const float*)d_in[0];
    const int*   eidx = (const int*)d_in[1];
    const float* W1  = (const float*)d_in[2];
    const float* as1 = (const float*)d_in[3];
    const float* ad1 = (const float*)d_in[4];
    const float* b1  = (const float*)d_in[5];
    const float* W2  = (const float*)d_in[6];
    const float* as2 = (const float*)d_in[7];
    const float* ad2 = (const float*)d_in[8];
    const float* b2  = (const float*)d_in[9];
    const float* Wm1 = (const float*)d_in[10];
    const float* bm1 = (const float*)d_in[11];
    const float* Wm2 = (const float*)d_in[12];
    const float* bm2 = (const float*)d_in[13];

    const int FIN  = 256;
    const int N    = in_sizes[0] / FIN;          // 50000
    const int E    = in_sizes[1] / 2;            // 600000
    const int C    = in_sizes[12] / HDIM;        // 20
    const int Etot = E + N;
    const int NH   = N * HDIM;

    const int* src = eidx;
    const int* dst = eidx + E;

    float*    B0     = (float*)d_ws;             // NH
    float*    B1     = B0 + NH;                  // NH
    float*    alphaS = B1 + NH;                  // N
    float*    alphaD = alphaS + N;               // N
    unsigned* maxkey = (unsigned*)(alphaD + N);  // N
    float*    denom  = (float*)(maxkey + N);     // N
    float*    wexp   = denom + N;                // Etot

    const int bigBlocks  = (N + 127) / 128;      // 4 waves x 32 rows
    const int smallBlocks = ((N + 15) / 16 + 3) / 4;
    const int edgeBlocks = (Etot + 255) / 256;
    const int edgeWaveBl = (Etot + 7) / 8;       // 8 waves / 256-thread block
    const int nodeWaveBl = (N + 7) / 8;
    const int initBlocks = (NH + 255) / 256;

    // ---- GAT layer 1: h1pre = x @ W1 --------------------------------------
    gemm_wmma_lds<256, 0, 0><<<bigBlocks, 128, 0, stream>>>(
        x, W1, nullptr, nullptr, B0, N);
    node_alphas<<<nodeWaveBl, 256, 0, stream>>>(B0, as1, ad1, alphaS, alphaD, N);
    gat_init<<<initBlocks, 256, 0, stream>>>(B1, maxkey, denom, N);
    edge_logit_max<<<edgeBlocks, 256, 0, stream>>>(src, dst, alphaS, alphaD, maxkey, E, N);
    edge_expsum<<<edgeBlocks, 256, 0, stream>>>(src, dst, alphaS, alphaD, maxkey, wexp, denom, E, N);
    edge_aggregate<<<edgeWaveBl, 256, 0, stream>>>(src, dst, B0, wexp, denom, B1, E, N);

    // ---- GAT layer 2: h2pre = relu(agg1 + b1) @ W2 (bias+relu fused PRE) --
    gemm_wmma_lds<128, 2, 0><<<bigBlocks, 128, 0, stream>>>(
        B1, W2, b1, nullptr, B0, N);
    node_alphas<<<nodeWaveBl, 256, 0, stream>>>(B0, as2, ad2, alphaS, alphaD, N);
    gat_init<<<initBlocks, 256, 0, stream>>>(B1, maxkey, denom, N);
    edge_logit_max<<<edgeBlocks, 256, 0, stream>>>(src, dst, alphaS, alphaD, maxkey, E, N);
    edge_expsum<<<edgeBlocks, 256, 0, stream>>>(src, dst, alphaS, alphaD, maxkey, wexp, denom, E, N);
    edge_aggregate<<<edgeWaveBl, 256, 0, stream>>>(src, dst, B0, wexp, denom, B1, E, N);

    // ---- MLP head: z = relu((agg2 + b2) @ Wm1 + bm1) ----------------------
    gemm_wmma_lds<128, 1, 1><<<bigBlocks, 128, 0, stream>>>(
        B1, Wm1, b2, bm1, B0, N);
    // out = sigmoid(z @ Wm2 + bm2), Ncols = 20 (masked via cndmask)
    gemm_wmma_small<<<smallBlocks, 128, 0, stream>>>(
        B0, Wm2, bm2, (float*)d_out, N, HDIM, C);
}